// OHEMLoss_91139206021276
// MI455X (gfx1250) — compile-verified
//
#include <hip/hip_runtime.h>
#include <stdint.h>

// Problem constants (match reference setup_inputs)
#define BB 8
#define CC 19
#define HH 512
#define WW 1024
#define HWN (HH * WW)            // 524288 pixels per batch
#define THRESH 0.7f
#define NMIN 10000
#define BLK 256
#define PPT 4                     // pixels per thread (float4)
#define BLOCKS_PER_BATCH (HWN / (BLK * PPT))   // 512

typedef float v4f __attribute__((ext_vector_type(4)));

__device__ __forceinline__ float fast_exp2(float x) {
#if __has_builtin(__builtin_amdgcn_exp2f)
  return __builtin_amdgcn_exp2f(x);
#else
  return exp2f(x);
#endif
}
__device__ __forceinline__ float fast_log2(float x) {
#if __has_builtin(__builtin_amdgcn_logf)
  return __builtin_amdgcn_logf(x);
#else
  return log2f(x);
#endif
}

// ---------------------------------------------------------------------------
// Zero the per-batch accumulators (ws is poisoned, must re-init every launch)
// ---------------------------------------------------------------------------
__global__ void ohem_init(float* __restrict__ bsum, unsigned* __restrict__ bcnt) {
  int i = threadIdx.x;
  if (i < BB) { bsum[i] = 0.0f; bcnt[i] = 0u; }
}

// ---------------------------------------------------------------------------
// Phase 1: per-pixel CE loss (logsumexp - x[target]) + per-batch
//          count(loss>0.7) / sum(loss>0.7) accumulation.
// One thread = 4 consecutive pixels; one block = 1024 pixels of one batch.
// Targets are async-staged to LDS (gfx1250 async vmem path) and consumed
// after the 19 non-temporal channel loads, hiding their latency.
// ---------------------------------------------------------------------------
__global__ __launch_bounds__(BLK) void ohem_phase1(
    const float* __restrict__ pred, const int* __restrict__ tgt,
    float* __restrict__ losses, float* __restrict__ bsum,
    unsigned* __restrict__ bcnt) {
  __shared__ int   s_tgt[BLK * PPT];
  __shared__ float rs[BLK];
  __shared__ int   rc[BLK];

  const int tid = threadIdx.x;
  const int b   = blockIdx.x / BLOCKS_PER_BATCH;
  const int blk = blockIdx.x % BLOCKS_PER_BATCH;
  const int p0  = (blk * BLK + tid) * PPT;        // pixel index within batch

  // --- async-stage this lane's 4 target indices into LDS (16B) -------------
  {
    const int* g = tgt + (size_t)b * HWN + p0;
    // Low 32 bits of a generic pointer into the LDS aperture are the LDS
    // byte offset (flat aperture mapping, CDNA5 ISA 10.2).
    uint32_t laddr = (uint32_t)(size_t)(&s_tgt[tid * PPT]);
    asm volatile("global_load_async_to_lds_b128 %0, %1, off"
                 :: "v"(laddr), "v"((uint64_t)(size_t)g)
                 : "memory");
  }

  // --- stream all 19 channels (non-temporal: 319MB read-once stream) -------
  const v4f* pp = (const v4f*)(pred + (size_t)b * CC * HWN + p0);
  v4f x[CC];
#pragma unroll
  for (int c = 0; c < CC; ++c)
    x[c] = __builtin_nontemporal_load(pp + (size_t)c * (HWN / PPT));

  // --- per-component max over channels -------------------------------------
  float m0 = x[0].x, m1 = x[0].y, m2 = x[0].z, m3 = x[0].w;
#pragma unroll
  for (int c = 1; c < CC; ++c) {
    m0 = fmaxf(m0, x[c].x); m1 = fmaxf(m1, x[c].y);
    m2 = fmaxf(m2, x[c].z); m3 = fmaxf(m3, x[c].w);
  }

  // --- sum of exp(x - m) via exp2 ------------------------------------------
  const float L2E = 1.44269504088896340736f;
  float s0 = 0.f, s1 = 0.f, s2 = 0.f, s3 = 0.f;
#pragma unroll
  for (int c = 0; c < CC; ++c) {
    s0 += fast_exp2((x[c].x - m0) * L2E);
    s1 += fast_exp2((x[c].y - m1) * L2E);
    s2 += fast_exp2((x[c].z - m2) * L2E);
    s3 += fast_exp2((x[c].w - m3) * L2E);
  }

  // --- consume async-staged targets ----------------------------------------
  asm volatile("s_wait_asynccnt 0x0" ::: "memory");
  const int t0 = s_tgt[tid * PPT + 0];
  const int t1 = s_tgt[tid * PPT + 1];
  const int t2 = s_tgt[tid * PPT + 2];
  const int t3 = s_tgt[tid * PPT + 3];

  float a0 = 0.f, a1 = 0.f, a2 = 0.f, a3 = 0.f;   // x[target]
#pragma unroll
  for (int c = 0; c < CC; ++c) {
    a0 = (t0 == c) ? x[c].x : a0;
    a1 = (t1 == c) ? x[c].y : a1;
    a2 = (t2 == c) ? x[c].z : a2;
    a3 = (t3 == c) ? x[c].w : a3;
  }

  const float LN2 = 0.69314718055994530942f;
  const float l0 = m0 + LN2 * fast_log2(s0) - a0;
  const float l1 = m1 + LN2 * fast_log2(s1) - a1;
  const float l2 = m2 + LN2 * fast_log2(s2) - a2;
  const float l3 = m3 + LN2 * fast_log2(s3) - a3;

  // --- store losses (scratch for rare top-k fallback; NT to spare L2) ------
  v4f lv; lv.x = l0; lv.y = l1; lv.z = l2; lv.w = l3;
  __builtin_nontemporal_store(lv, (v4f*)(losses + (size_t)b * HWN + p0));

  // --- accumulate count/sum of losses above threshold ----------------------
  float ls = 0.f; int lc = 0;
  if (l0 > THRESH) { ls += l0; ++lc; }
  if (l1 > THRESH) { ls += l1; ++lc; }
  if (l2 > THRESH) { ls += l2; ++lc; }
  if (l3 > THRESH) { ls += l3; ++lc; }

  rs[tid] = ls; rc[tid] = lc;
  __syncthreads();
#pragma unroll
  for (int off = BLK / 2; off > 0; off >>= 1) {
    if (tid < off) { rs[tid] += rs[tid + off]; rc[tid] += rc[tid + off]; }
    __syncthreads();
  }
  if (tid == 0) {
    atomicAdd(&bsum[b], rs[0]);
    atomicAdd(&bcnt[b], (unsigned)rc[0]);
  }
}

// ---------------------------------------------------------------------------
// Phase 2: per-batch mean of kept losses, then batch average.
// Common case: count(>0.7) >= 10000  ->  mean = sum/count (O(1)).
// Fallback: exact top-10000 via 32-step threshold bisection over the stored
// losses (counts <= 2^24 are exact in f32), ties handled analytically.
// ---------------------------------------------------------------------------
__global__ __launch_bounds__(BLK) void ohem_phase2(
    const float* __restrict__ losses, const float* __restrict__ bsum,
    const unsigned* __restrict__ bcnt, float* __restrict__ out) {
  __shared__ float red[BLK];
  const int tid = threadIdx.x;
  float total = 0.0f;

  for (int b = 0; b < BB; ++b) {
    const unsigned c0 = bcnt[b];
    const float    s0 = bsum[b];
    float mean;
    if (c0 >= (unsigned)NMIN) {
      mean = s0 / (float)c0;           // kept set == {loss > THRESH}
    } else {
      const float* Lb = losses + (size_t)b * HWN;
      float lo = 0.0f, hi = THRESH;    // CE loss >= 0; k-th value in [0,THRESH]
      for (int it = 0; it < 32; ++it) {
        const float mid = 0.5f * (lo + hi);
        float cnt = 0.0f;
        for (int i = tid; i < HWN; i += BLK) cnt += (Lb[i] > mid) ? 1.0f : 0.0f;
        red[tid] = cnt; __syncthreads();
        for (int off = BLK / 2; off > 0; off >>= 1) {
          if (tid < off) red[tid] += red[tid + off];
          __syncthreads();
        }
        cnt = red[0]; __syncthreads();
        if (cnt >= (float)NMIN) lo = mid; else hi = mid;
      }
      // count & sum strictly above lo
      float cnt = 0.0f, sum = 0.0f;
      for (int i = tid; i < HWN; i += BLK) {
        const float v = Lb[i];
        if (v > lo) { cnt += 1.0f; sum += v; }
      }
      red[tid] = cnt; __syncthreads();
      for (int off = BLK / 2; off > 0; off >>= 1) {
        if (tid < off) red[tid] += red[tid + off];
        __syncthreads();
      }
      const float c1 = red[0]; __syncthreads();
      red[tid] = sum; __syncthreads();
      for (int off = BLK / 2; off > 0; off >>= 1) {
        if (tid < off) red[tid] += red[tid + off];
        __syncthreads();
      }
      const float s1 = red[0]; __syncthreads();
      mean = (s1 + ((float)NMIN - c1) * lo) / (float)NMIN;
    }
    total += mean;
  }
  if (tid == 0) out[0] = total / (float)BB;
}

// ---------------------------------------------------------------------------
extern "C" void kernel_launch(void* const* d_in, const int* in_sizes, int n_in,
                              void* d_out, int out_size, void* d_ws,
                              size_t ws_size, hipStream_t stream) {
  const float* pred = (const float*)d_in[0];
  const int*   tgt  = (const int*)d_in[1];
  float*       out  = (float*)d_out;

  char* ws = (char*)d_ws;
  float*    bsum   = (float*)ws;               // 8 floats
  unsigned* bcnt   = (unsigned*)(ws + 64);     // 8 uints
  float*    losses = (float*)(ws + 256);       // B*HW floats = 16 MB

  ohem_init<<<1, 32, 0, stream>>>(bsum, bcnt);
  ohem_phase1<<<dim3(BB * BLOCKS_PER_BATCH), dim3(BLK), 0, stream>>>(
      pred, tgt, losses, bsum, bcnt);
  ohem_phase2<<<1, BLK, 0, stream>>>(losses, bsum, bcnt, out);
}